// GINLayer_73512660239033
// MI455X (gfx1250) — compile-verified
//
#include <hip/hip_runtime.h>

// GIN layer for MI455X (gfx1250):
//  k1: X_agg = X                      (coalesced float4 copy into d_out)
//  k2: X_agg[a]+=X[b]; X_agg[b]+=X[a] (edge-parallel float4 atomic scatter, L2-resident)
//  k3: fused 2-layer MLP with V_WMMA_F32_16X16X4_F32 (exact f32, matrix pipe),
//      weights in LDS, inter-layer tile staged in wave-private LDS, in-place on d_out.

#define N_NODES 100000
#define N_EDGES 1000000
#define D_FEAT  64

typedef __attribute__((ext_vector_type(2))) float v2f;
typedef __attribute__((ext_vector_type(8))) float v8f;

// ---------------- kernel 1: X_agg = X ----------------
__global__ void gin_copy(const float* __restrict__ X, float* __restrict__ Xagg) {
  size_t i = (size_t)blockIdx.x * blockDim.x + threadIdx.x;
  const size_t n4 = (size_t)N_NODES * D_FEAT / 4;
  if (i < n4) ((float4*)Xagg)[i] = ((const float4*)X)[i];
}

// ---------------- kernel 2: undirected edge scatter-add ----------------
// 16 threads per edge, 4 floats each (float4 loads, scalar f32 atomics).
__global__ void gin_scatter(const float* __restrict__ X,
                            const int*   __restrict__ ra,
                            const int*   __restrict__ rb,
                            float*       __restrict__ Xagg) {
  size_t idx = (size_t)blockIdx.x * blockDim.x + threadIdx.x;
  size_t e = idx >> 4;
  if (e >= N_EDGES) return;
  int f = (int)(idx & 15) * 4;
  int a = ra[e];
  int b = rb[e];
  float4 xb = *(const float4*)(X + (size_t)b * D_FEAT + f);
  float4 xa = *(const float4*)(X + (size_t)a * D_FEAT + f);
  float* pa = Xagg + (size_t)a * D_FEAT + f;
  float* pb = Xagg + (size_t)b * D_FEAT + f;
  atomicAdd(pa + 0, xb.x); atomicAdd(pa + 1, xb.y);
  atomicAdd(pa + 2, xb.z); atomicAdd(pa + 3, xb.w);
  atomicAdd(pb + 0, xa.x); atomicAdd(pb + 1, xa.y);
  atomicAdd(pb + 2, xa.z); atomicAdd(pb + 3, xa.w);
}

// ---------------- kernel 3: fused MLP, in-place on X_agg (== d_out) ----------------
// 256 threads = 8 wave32s; each wave owns one 16-row tile and does the full
// 16x64 output via 4 N-blocks x 16 K-steps of V_WMMA_F32_16X16X4_F32 per layer.
__launch_bounds__(256)
__global__ void gin_mlp(float* __restrict__ Xagg,
                        const float* __restrict__ Wh, const float* __restrict__ bh,
                        const float* __restrict__ Wo, const float* __restrict__ bo) {
  __shared__ float sWh[D_FEAT * D_FEAT];        // 16 KB
  __shared__ float sWo[D_FEAT * D_FEAT];        // 16 KB
  __shared__ float sH[8 * 16 * D_FEAT];         // 32 KB, per-wave 16x64 hidden tile

  const int tid = threadIdx.x;
  for (int i = tid; i < D_FEAT * D_FEAT; i += 256) {
    sWh[i] = Wh[i];
    sWo[i] = Wo[i];
  }
  __syncthreads();

  const int wave = tid >> 5;
  const int lane = tid & 31;
  const int half = lane >> 4;   // A layout: lanes 0-15 hold K={0,1}, 16-31 hold K={2,3}
  const int lrow = lane & 15;
  const int tile = blockIdx.x * 8 + wave;
  if (tile * 16 >= N_NODES) return;             // wave-uniform: EXEC stays all-1s
  const int rowBase = tile * 16;

  float biasH[4], biasO[4];
#pragma unroll
  for (int nb = 0; nb < 4; ++nb) {
    biasH[nb] = bh[nb * 16 + lrow];             // bias depends on column only
    biasO[nb] = bo[nb * 16 + lrow];
  }

  // ---- layer 1: H = relu(Xagg_tile @ Wh + bh) ----
  v8f acc[4] = {};
  const float* arow = Xagg + (size_t)(rowBase + lrow) * D_FEAT + half * 2;
#pragma unroll
  for (int kb = 0; kb < 16; ++kb) {
    v2f a;
    a.x = arow[kb * 4 + 0];                     // A[M=lrow][k0], A[M=lrow][k0+1]
    a.y = arow[kb * 4 + 1];
    const int k0 = kb * 4 + half * 2;
#pragma unroll
    for (int nb = 0; nb < 4; ++nb) {
      v2f b;
      b.x = sWh[(k0 + 0) * D_FEAT + nb * 16 + lrow];   // B row K=k0,   N=col
      b.y = sWh[(k0 + 1) * D_FEAT + nb * 16 + lrow];   // B row K=k0+1, N=col
      acc[nb] = __builtin_amdgcn_wmma_f32_16x16x4_f32(
          false, a, false, b, (short)0, acc[nb], false, false);
    }
  }

  // bias+relu, stage hidden tile in wave-private LDS (C layout: M = half*8 + i)
  float* myH = sH + wave * (16 * D_FEAT);
#pragma unroll
  for (int nb = 0; nb < 4; ++nb) {
#pragma unroll
    for (int i = 0; i < 8; ++i) {
      float v = acc[nb][i] + biasH[nb];
      v = v > 0.0f ? v : 0.0f;
      myH[(half * 8 + i) * D_FEAT + nb * 16 + lrow] = v;
    }
  }
  // DS ops from the same wave are in-order (DScnt) -> no barrier needed.

  // ---- layer 2: O = relu(H @ Wo + bo) ----
  v8f acc2[4] = {};
  const float* hrow = myH + lrow * D_FEAT + half * 2;
#pragma unroll
  for (int kb = 0; kb < 16; ++kb) {
    v2f a;
    a.x = hrow[kb * 4 + 0];
    a.y = hrow[kb * 4 + 1];
    const int k0 = kb * 4 + half * 2;
#pragma unroll
    for (int nb = 0; nb < 4; ++nb) {
      v2f b;
      b.x = sWo[(k0 + 0) * D_FEAT + nb * 16 + lrow];
      b.y = sWo[(k0 + 1) * D_FEAT + nb * 16 + lrow];
      acc2[nb] = __builtin_amdgcn_wmma_f32_16x16x4_f32(
          false, a, false, b, (short)0, acc2[nb], false, false);
    }
  }

  // epilogue: bias+relu, write this wave's own 16 rows back (in-place is safe:
  // all reads of these rows happened in layer 1 above, same wave).
#pragma unroll
  for (int nb = 0; nb < 4; ++nb) {
#pragma unroll
    for (int i = 0; i < 8; ++i) {
      float v = acc2[nb][i] + biasO[nb];
      v = v > 0.0f ? v : 0.0f;
      Xagg[(size_t)(rowBase + half * 8 + i) * D_FEAT + nb * 16 + lrow] = v;
    }
  }
}

extern "C" void kernel_launch(void* const* d_in, const int* in_sizes, int n_in,
                              void* d_out, int out_size, void* d_ws, size_t ws_size,
                              hipStream_t stream) {
  const float* X     = (const float*)d_in[0];
  const int*   ref_a = (const int*)  d_in[1];
  const int*   ref_b = (const int*)  d_in[2];
  // d_in[3] v_map, d_in[4] v_count: unused by the reference computation
  const float* Wh    = (const float*)d_in[5];
  const float* bh    = (const float*)d_in[6];
  const float* Wo    = (const float*)d_in[7];
  const float* bo    = (const float*)d_in[8];
  float* out = (float*)d_out;   // doubles as X_agg scratch (fully rewritten each call)

  // k1: copy X -> out (1.6M float4)
  {
    size_t n4 = (size_t)N_NODES * D_FEAT / 4;
    int blocks = (int)((n4 + 255) / 256);
    gin_copy<<<blocks, 256, 0, stream>>>(X, out);
  }
  // k2: edge scatter, 16 threads/edge
  {
    size_t threads = (size_t)N_EDGES * 16;
    int blocks = (int)((threads + 255) / 256);
    gin_scatter<<<blocks, 256, 0, stream>>>(X, ref_a, ref_b, out);
  }
  // k3: fused WMMA MLP, 8 tiles (waves) per 256-thread block
  {
    int tiles = (N_NODES + 15) / 16;          // 6250
    int blocks = (tiles + 7) / 8;             // 782
    gin_mlp<<<blocks, 256, 0, stream>>>(out, Wh, bh, Wo, bo);
  }
}